// HypAgg_36799279793047
// MI455X (gfx1250) — compile-verified
//
#include <hip/hip_runtime.h>
#include <cstdint>

// ---------------------------------------------------------------------------
// HypAgg: support = segment_sum(adj_vals * logmap0(x)[adj_cols], adj_rows)
// N=50000 nodes, D=128 feats, E=800000 edges, c=1. Memory/atomic bound:
// ~0.2 GFLOP vs ~900MB of (mostly L2-resident) traffic -> no WMMA; optimize
// data movement: wave-per-row float4 loads, async global->LDS edge staging,
// hardware f32 global atomics (inline asm, device scope).
// ---------------------------------------------------------------------------

#define D_FEAT 128
#define TILE   256   // edges per block
#define WAVE   32

#if __has_builtin(__builtin_amdgcn_global_load_async_to_lds_b32)
#define HAVE_ASYNC_LDS 1
#else
#define HAVE_ASYNC_LDS 0
#endif

// clang's builtin signature: (int AS1*, int AS3*, imm offset, imm cpol)
#define GLOBAL_I(p) ((__attribute__((address_space(1))) int*)(uintptr_t)(p))
#define LDS_I(p)    ((__attribute__((address_space(3))) int*)(uintptr_t)(p))

#if HAVE_ASYNC_LDS
__device__ __forceinline__ void wait_async0() {
#if __has_builtin(__builtin_amdgcn_s_wait_asynccnt)
  __builtin_amdgcn_s_wait_asynccnt(0);
#else
  asm volatile("s_wait_asynccnt 0" ::: "memory");
#endif
}
#endif

// Guaranteed hardware FP atomic: no-return global_atomic_add_f32 at device
// scope (output rows are hit by atomics from every WGP).
__device__ __forceinline__ void atomic_add_f32(float* p, float v) {
  asm volatile("global_atomic_add_f32 %0, %1, off scope:SCOPE_DEV"
               :: "v"(p), "v"(v) : "memory");
}

__device__ __forceinline__ float wave_reduce_add(float v) {
#pragma unroll
  for (int off = 16; off > 0; off >>= 1) v += __shfl_xor(v, off, 32);
  return v;
}

// scale = artanh(clip(norm,..,1-eps)) / max(norm, 1e-15), c = 1
__device__ __forceinline__ float logmap0_scale(float ss) {
  float norm = sqrtf(ss);
  float xn = fmaxf(norm, 1e-15f);
  float u = fminf(xn, 0.99999994f);               // 1 - 2^-24 (f32 clip)
  float at = 0.5f * (log1pf(u) - log1pf(-u));     // artanh, accurate small u
  return at / xn;
}

// ---------------------------------------------------------------------------
// HOT KERNEL (defined first so the disasm snippet shows it).
// Block = one 256-edge tile. Stage (row,col,val) into LDS via async-to-LDS,
// then each wave handles 32 edges: broadcast metadata by shuffle, float4
// gather of the 128-feat tangent row, 4 f32 global atomics per lane.
// ---------------------------------------------------------------------------
__global__ void edge_scatter_pre(const float* __restrict__ xt,
                                 const int* __restrict__ rows,
                                 const int* __restrict__ cols,
                                 const float* __restrict__ vals,
                                 float* __restrict__ out, int n_edges) {
  __shared__ int   s_rows[TILE];
  __shared__ int   s_cols[TILE];
  __shared__ float s_vals[TILE];

  const int t    = threadIdx.x;
  const int base = blockIdx.x * TILE;
  const int e    = base + t;

#if HAVE_ASYNC_LDS
  if (e < n_edges) {
    __builtin_amdgcn_global_load_async_to_lds_b32(GLOBAL_I(rows + e), LDS_I(&s_rows[t]), 0, 0);
    __builtin_amdgcn_global_load_async_to_lds_b32(GLOBAL_I(cols + e), LDS_I(&s_cols[t]), 0, 0);
    __builtin_amdgcn_global_load_async_to_lds_b32(GLOBAL_I(vals + e), LDS_I(&s_vals[t]), 0, 0);
  }
  wait_async0();
  __syncthreads();
#else
  if (e < n_edges) {
    s_rows[t] = rows[e];
    s_cols[t] = cols[e];
    s_vals[t] = vals[e];
  }
  __syncthreads();
#endif

  const int lane = t & 31;
  const int wv   = t >> 5;
  const int j0   = wv * WAVE;

  const int myE = base + j0 + lane;
  int   r = (myE < n_edges) ? s_rows[j0 + lane] : -1;
  int   c = (myE < n_edges) ? s_cols[j0 + lane] : 0;
  float w = (myE < n_edges) ? s_vals[j0 + lane] : 0.f;

  const float4* xt4 = (const float4*)xt;

#pragma unroll 4
  for (int j = 0; j < WAVE; ++j) {
    int rr = __shfl(r, j, 32);        // wave-uniform
    if (rr < 0) continue;
    int   cc = __shfl(c, j, 32);
    float ww = __shfl(w, j, 32);
    float4 v = xt4[cc * (D_FEAT / 4) + lane];   // 512B coalesced, L2-resident
    float* dst = out + (size_t)rr * D_FEAT + lane * 4;
    atomic_add_f32(dst + 0, v.x * ww);
    atomic_add_f32(dst + 1, v.y * ww);
    atomic_add_f32(dst + 2, v.z * ww);
    atomic_add_f32(dst + 3, v.w * ww);
  }
}

// One wave per node: lane l holds feats [4l, 4l+4)
__global__ void logmap0_kernel(const float* __restrict__ x,
                               float* __restrict__ xt, int n_nodes) {
  int gid  = blockIdx.x * blockDim.x + threadIdx.x;
  int node = gid >> 5;
  int lane = gid & 31;
  if (node >= n_nodes) return;
  const float4* row = (const float4*)(x + (size_t)node * D_FEAT);
  float4 v = row[lane];
  float ss = wave_reduce_add(v.x * v.x + v.y * v.y + v.z * v.z + v.w * v.w);
  float s = logmap0_scale(ss);
  float4 o{v.x * s, v.y * s, v.z * s, v.w * s};
  ((float4*)(xt + (size_t)node * D_FEAT))[lane] = o;
}

// Fallback if workspace is too small: recompute logmap0 per edge post-gather.
__global__ void edge_scatter_fused(const float* __restrict__ x,
                                   const int* __restrict__ rows,
                                   const int* __restrict__ cols,
                                   const float* __restrict__ vals,
                                   float* __restrict__ out, int n_edges) {
  __shared__ int   s_rows[TILE];
  __shared__ int   s_cols[TILE];
  __shared__ float s_vals[TILE];

  const int t    = threadIdx.x;
  const int base = blockIdx.x * TILE;
  const int e    = base + t;

  if (e < n_edges) {
    s_rows[t] = rows[e];
    s_cols[t] = cols[e];
    s_vals[t] = vals[e];
  }
  __syncthreads();

  const int lane = t & 31;
  const int j0   = (t >> 5) * WAVE;
  const int myE  = base + j0 + lane;
  int   r = (myE < n_edges) ? s_rows[j0 + lane] : -1;
  int   c = (myE < n_edges) ? s_cols[j0 + lane] : 0;
  float w = (myE < n_edges) ? s_vals[j0 + lane] : 0.f;

  const float4* x4 = (const float4*)x;
  for (int j = 0; j < WAVE; ++j) {
    int rr = __shfl(r, j, 32);
    if (rr < 0) continue;
    int   cc = __shfl(c, j, 32);
    float ww = __shfl(w, j, 32);
    float4 v = x4[cc * (D_FEAT / 4) + lane];
    float ss = wave_reduce_add(v.x * v.x + v.y * v.y + v.z * v.z + v.w * v.w);
    float s = ww * logmap0_scale(ss);
    float* dst = out + (size_t)rr * D_FEAT + lane * 4;
    atomic_add_f32(dst + 0, v.x * s);
    atomic_add_f32(dst + 1, v.y * s);
    atomic_add_f32(dst + 2, v.z * s);
    atomic_add_f32(dst + 3, v.w * s);
  }
}

__global__ void zero_f4(float4* __restrict__ out, int n4) {
  int i = blockIdx.x * blockDim.x + threadIdx.x;
  if (i < n4) out[i] = float4{0.f, 0.f, 0.f, 0.f};
}

extern "C" void kernel_launch(void* const* d_in, const int* in_sizes, int n_in,
                              void* d_out, int out_size, void* d_ws, size_t ws_size,
                              hipStream_t stream) {
  const float* x    = (const float*)d_in[0];
  const int*   rows = (const int*)d_in[1];
  const int*   cols = (const int*)d_in[2];
  const float* vals = (const float*)d_in[3];
  float*       out  = (float*)d_out;

  const int n_nodes = in_sizes[0] / D_FEAT;
  const int n_edges = in_sizes[1];

  // 1) zero the output (segment_sum accumulator)
  const int n4 = out_size / 4;  // D_FEAT==128 -> divisible by 4
  zero_f4<<<(n4 + 255) / 256, 256, 0, stream>>>((float4*)out, n4);

  const int edge_blocks = (n_edges + TILE - 1) / TILE;
  const size_t ws_needed = (size_t)in_sizes[0] * sizeof(float);

  if (ws_size >= ws_needed) {
    // 2) precompute tangent vectors once (25.6MB, fits L2)
    float* xt = (float*)d_ws;
    const int threads = 256;
    const int blocks = (n_nodes * WAVE + threads - 1) / threads;
    logmap0_kernel<<<blocks, threads, 0, stream>>>(x, xt, n_nodes);
    // 3) gather + scatter-add
    edge_scatter_pre<<<edge_blocks, TILE, 0, stream>>>(
        xt, rows, cols, vals, out, n_edges);
  } else {
    // fused fallback: recompute logmap0 per edge after the gather
    edge_scatter_fused<<<edge_blocks, TILE, 0, stream>>>(
        x, rows, cols, vals, out, n_edges);
  }
}